// SELayer_16930761081313
// MI455X (gfx1250) — compile-verified
//
#include <hip/hip_runtime.h>
#include <hip/hip_bf16.h>

// ---- shapes from the reference ----
#define B        32
#define C        512
#define H        56
#define W_DIM    56
#define MODALITY 8
#define HW       (H * W_DIM)          // 3136
#define HW4      (HW / 4)             // 784 float4 per (b,c) plane
#define PLANES   (B * C)              // 16384
#define TOTAL4   ((size_t)PLANES * HW4)

typedef __attribute__((ext_vector_type(2))) float v2f;
typedef __attribute__((ext_vector_type(4))) float v4f;
typedef __attribute__((ext_vector_type(8))) float v8f;

// ---------------------------------------------------------------------------
// Kernel 1: per-(b,c) spatial mean.  One block per plane, 256 threads,
// fully coalesced b128 loads (784 float4 / plane).
// ---------------------------------------------------------------------------
__global__ __launch_bounds__(256) void se_mean(const v4f* __restrict__ x4,
                                               float* __restrict__ y) {
  const int plane = blockIdx.x;                 // b*C + c
  const v4f* p = x4 + (size_t)plane * HW4;

  float sum = 0.f;
  for (int i = threadIdx.x; i < HW4; i += 256) {
    v4f v = p[i];
    sum += (v.x + v.y) + (v.z + v.w);
  }

  // wave32 reduce
  #pragma unroll
  for (int off = 16; off > 0; off >>= 1)
    sum += __shfl_down(sum, off, 32);

  __shared__ float smem[8];
  const int lane = threadIdx.x & 31;
  const int wv   = threadIdx.x >> 5;
  if (lane == 0) smem[wv] = sum;
  __syncthreads();
  if (threadIdx.x == 0) {
    float t = 0.f;
    #pragma unroll
    for (int w = 0; w < 8; ++w) t += smem[w];
    y[plane] = t * (1.0f / (float)HW);
  }
}

// ---------------------------------------------------------------------------
// Kernel 2: gate = sigmoid(relu(y @ W^T + bias)) via V_WMMA_F32_16X16X4_F32.
// One block, 2 waves. Wave w handles batch rows [16w, 16w+16).
// A (16x4 f32) lane layout (ISA 7.12.2): lanes 0-15 row M=lane hold K=k0,k0+1;
// lanes 16-31 row M=lane-16 hold K=k0+2,k0+3.  B (4x16) mirrors over N.
// C/D: VGPR v <-> row v + 8*(lane>=16), N = lane&15.  N padded 8->16 w/ zeros.
// ---------------------------------------------------------------------------
__global__ __launch_bounds__(64) void se_gate_wmma(const float* __restrict__ y,
                                                   const float* __restrict__ Wm,
                                                   const float* __restrict__ bias,
                                                   float* __restrict__ gate) {
  const int lane  = threadIdx.x & 31;
  const int wave  = threadIdx.x >> 5;           // 0 or 1
  const int mrow  = (lane & 15) + wave * 16;    // A row fed by this lane
  const int khalf = (lane >> 4) << 1;           // 0 for lanes 0-15, 2 for 16-31
  const int ncol  = lane & 15;                  // B column fed by this lane
  const bool nvalid = (ncol < MODALITY);

  const float* yrow = y + (size_t)mrow * C;
  const float* wrow = Wm + (size_t)(nvalid ? ncol : 0) * C;  // safe addr; zeroed below

  v8f acc = {};
  for (int k0 = 0; k0 < C; k0 += 4) {
    v2f a, b;
    a.x = yrow[k0 + khalf + 0];
    a.y = yrow[k0 + khalf + 1];
    float b0 = wrow[k0 + khalf + 0];
    float b1 = wrow[k0 + khalf + 1];
    b.x = nvalid ? b0 : 0.f;                    // lane-uniform select, EXEC stays all-1s
    b.y = nvalid ? b1 : 0.f;
    acc = __builtin_amdgcn_wmma_f32_16x16x4_f32(
        /*neg_a=*/false, a, /*neg_b=*/false, b,
        /*c_mod=*/(short)0, acc, /*reuse_a=*/false, /*reuse_b=*/false);
  }

  if (nvalid) {
    const float bn   = bias[ncol];
    const int bbase  = wave * 16 + ((lane >> 4) << 3);  // +8 for lanes 16-31
    #pragma unroll
    for (int v = 0; v < 8; ++v) {
      float z = acc[v] + bn;
      z = fmaxf(z, 0.f);                        // relu
      gate[(bbase + v) * MODALITY + ncol] = 1.0f / (1.0f + __expf(-z));
    }
  }
}

// ---------------------------------------------------------------------------
// Kernel 3: out = x * gate[b, c>>6].  One float4 per thread, streaming NT
// store (output is write-once; keep it out of the caches).
// ---------------------------------------------------------------------------
__global__ __launch_bounds__(256) void se_scale(const v4f* __restrict__ x4,
                                                const float* __restrict__ gate,
                                                v4f* __restrict__ out4) {
  const size_t i = (size_t)blockIdx.x * 256 + threadIdx.x;
  if (i >= TOTAL4) return;
  const int plane = (int)(i / HW4);             // b*C + c
  const int b = plane >> 9;                     // / C
  const int c = plane & (C - 1);
  const float s = gate[b * MODALITY + (c >> 6)];

  v4f v = x4[i];
  v.x *= s; v.y *= s; v.z *= s; v.w *= s;
  __builtin_nontemporal_store(v, &out4[i]);
}

// ---------------------------------------------------------------------------
extern "C" void kernel_launch(void* const* d_in, const int* in_sizes, int n_in,
                              void* d_out, int out_size, void* d_ws, size_t ws_size,
                              hipStream_t stream) {
  (void)in_sizes; (void)n_in; (void)out_size; (void)ws_size;

  const float* x    = (const float*)d_in[0];    // (32, 512, 56, 56)
  const float* Wm   = (const float*)d_in[1];    // (8, 512)
  const float* bias = (const float*)d_in[2];    // (8,)
  float* out        = (float*)d_out;

  // workspace: y (32*512 floats) then gate (32*8 floats)
  float* y    = (float*)d_ws;
  float* gate = y + B * C;

  // Pass 1: spatial means
  se_mean<<<PLANES, 256, 0, stream>>>((const v4f*)x, y);

  // Pass 2: tiny GEMM + activation on the matrix core
  se_gate_wmma<<<1, 64, 0, stream>>>(y, Wm, bias, gate);

  // Pass 3: broadcast scale (streaming)
  const int nblk = (int)((TOTAL4 + 255) / 256);
  se_scale<<<nblk, 256, 0, stream>>>((const v4f*)x, gate, (v4f*)out);
}